// RegpPooling2D_41738492183256
// MI455X (gfx1250) — compile-verified
//
#include <hip/hip_runtime.h>

// RegpPooling2D for x:(8,64,130,130) f32 -> p:(8,64,128,128) f32.
// VALU-bound stencil; staged through LDS via CDNA5 Tensor Data Mover.

#define H_IN 130
#define W_IN 130
#define HO 128
#define WO 128
#define TILE_ROWS 32              // output rows per block
#define IN_ROWS (TILE_ROWS + 2)   // 34 input rows per block
#define LDS_ELEMS (IN_ROWS * W_IN) // 4420 floats, contiguous in global memory

typedef __attribute__((ext_vector_type(4))) unsigned int u32x4;
typedef __attribute__((ext_vector_type(4))) int i32x4;
typedef __attribute__((ext_vector_type(8))) int i32x8;

// One output pixel from its 3x3 window. Exactly replicates the reference:
// for each of the 9 window entries c, count neighbors (in the zero-padded
// window) with |v-c| <= 0.3|c|, minus self; argmax with tie-average update.
// Out-of-window cells are zeros: |0-c| <= 0.3|c|  <=>  |c| <= 0.3|c|.
__device__ __forceinline__ float regp_pixel(
    float w00, float w01, float w02,
    float w10, float w11, float w12,
    float w20, float w21, float w22) {
  const float w[3][3] = {{w00, w01, w02}, {w10, w11, w12}, {w20, w21, w22}};
  float p = 0.0f;
  float maxc = 0.0f;
#pragma unroll
  for (int ni = 0; ni < 3; ++ni) {
#pragma unroll
    for (int nj = 0; nj < 3; ++nj) {
      const float c = w[ni][nj];
      const float tol = 0.3f * __builtin_fabsf(c);
      float cnt = -1.0f;  // subtract the self-match
#pragma unroll
      for (int di = -1; di <= 1; ++di) {
#pragma unroll
        for (int dj = -1; dj <= 1; ++dj) {
          const int ri = ni + di, rj = nj + dj;
          if (ri >= 0 && ri < 3 && rj >= 0 && rj < 3) {
            cnt += (__builtin_fabsf(w[ri][rj] - c) <= tol) ? 1.0f : 0.0f;
          }
        }
      }
      // cells outside the 3x3 window are zero-padded
      const int noob = 9 - (ni == 1 ? 3 : 2) * (nj == 1 ? 3 : 2);
      if (noob) cnt += (__builtin_fabsf(c) <= tol) ? (float)noob : 0.0f;
      if (cnt > maxc) { p = c; maxc = cnt; }
      if (cnt == maxc) p = 0.5f * (c + p);
    }
  }
  return p;
}

__global__ __launch_bounds__(256) void regp_pool_kernel(
    const float* __restrict__ x, float* __restrict__ out) {
  __shared__ float lds[LDS_ELEMS];

  const int bc = (int)blockIdx.x >> 2;   // which (b,c) plane, 0..511
  const int tile = (int)blockIdx.x & 3;  // which 32-row band
  const int row0 = tile * TILE_ROWS;
  // The 34x130 input tile is contiguous in global memory: flat copy.
  const float* src = x + (size_t)bc * (H_IN * W_IN) + (size_t)row0 * W_IN;

#if defined(__gfx1250__) && __has_builtin(__builtin_amdgcn_tensor_load_to_lds)
  // Tensor Data Mover: one instruction DMAs the whole tile into LDS.
  if (threadIdx.x == 0) {  // one TDM issue per block (wave0; TDM ignores EXEC)
    const unsigned long long ga = (unsigned long long)(size_t)src;
    u32x4 g0;
    g0.x = 1u;                               // count=1, user descriptor
    g0.y = (unsigned)(size_t)(&lds[0]);      // lds_addr (byte offset)
    g0.z = (unsigned)(ga & 0xFFFFFFFFull);   // global_addr[31:0]
    g0.w = (unsigned)((ga >> 32) & 0x01FFFFFFull) | 0x80000000u;  // type=2
    i32x8 g1;
    g1[0] = 0x00020000;                                   // data_size=2 (4B)
    g1[1] = (int)(((unsigned)LDS_ELEMS & 0xFFFFu) << 16); // tensor_dim0 lo
    g1[2] = (int)((((unsigned)LDS_ELEMS >> 16) & 0xFFFFu) // tensor_dim0 hi
                  | (1u << 16));                          // tensor_dim1 = 1
    g1[3] = (int)(((unsigned)LDS_ELEMS & 0xFFFFu) << 16); // tile_dim0 = 4420
    g1[4] = 1;                                            // tile_dim1=1, tile_dim2=0
    g1[5] = LDS_ELEMS;                                    // tensor_dim0_stride
    g1[6] = 0;
    g1[7] = 0;
    const i32x4 z4 = {0, 0, 0, 0};
#if defined(__clang_major__) && (__clang_major__ >= 23)
    const i32x8 z8 = {0, 0, 0, 0, 0, 0, 0, 0};
    __builtin_amdgcn_tensor_load_to_lds(g0, g1, z4, z4, z8, 0);
#else
    __builtin_amdgcn_tensor_load_to_lds(g0, g1, z4, z4, 0);
#endif
  }
#if __has_builtin(__builtin_amdgcn_s_wait_tensorcnt)
  __builtin_amdgcn_s_wait_tensorcnt(0);  // no-op for waves that issued nothing
#endif
#else
  // Fallback: cooperative coalesced copy (tile is contiguous).
  for (int i = (int)threadIdx.x; i < LDS_ELEMS; i += 256) lds[i] = src[i];
#endif
  __syncthreads();

  // 256 threads: col = tid&127, half = tid>>7; each thread computes 16 rows
  // at one column with a rolling 3-row register window (3 LDS reads/pixel).
  const int col = (int)threadIdx.x & 127;
  const int half = (int)threadIdx.x >> 7;
  const int rbase = half * 16;  // local input row of first window row

  const float* l0 = &lds[rbase * W_IN + col];
  float w00 = l0[0],          w01 = l0[1],            w02 = l0[2];
  float w10 = l0[W_IN],       w11 = l0[W_IN + 1],     w12 = l0[W_IN + 2];
  float w20 = l0[2 * W_IN],   w21 = l0[2 * W_IN + 1], w22 = l0[2 * W_IN + 2];

  float* dst = out + (size_t)bc * (HO * WO) + (size_t)(row0 + rbase) * WO + col;

#pragma unroll 4
  for (int r = 0; r < 16; ++r) {
    dst[(size_t)r * WO] = regp_pixel(w00, w01, w02, w10, w11, w12, w20, w21, w22);
    if (r < 15) {
      const float* ln = &lds[(rbase + r + 3) * W_IN + col];
      w00 = w10; w01 = w11; w02 = w12;
      w10 = w20; w11 = w21; w12 = w22;
      w20 = ln[0]; w21 = ln[1]; w22 = ln[2];
    }
  }
}

extern "C" void kernel_launch(void* const* d_in, const int* in_sizes, int n_in,
                              void* d_out, int out_size, void* d_ws, size_t ws_size,
                              hipStream_t stream) {
  (void)in_sizes; (void)n_in; (void)d_ws; (void)ws_size; (void)out_size;
  const float* x = (const float*)d_in[0];
  float* out = (float*)d_out;
  // 8*64 planes * 4 row-bands = 2048 blocks, 256 threads (8 wave32 waves)
  dim3 grid(8 * 64 * 4), block(256);
  regp_pool_kernel<<<grid, block, 0, stream>>>(x, out);
}